// NeuralAdditiveCNN_53867479826735
// MI455X (gfx1250) — compile-verified
//
#include <hip/hip_runtime.h>
#include <math.h>

typedef __attribute__((ext_vector_type(2))) float v2f;
typedef __attribute__((ext_vector_type(8))) float v8f;

// ---------------------------------------------------------------------------
// Stage A: fused  bilinear-downsample (=2x2 avg) + conv1(1->8) + conv2(8->8)
//          + relu + maxpool2   ->  p1 [256, 8, 48, 96]
// Tile: 16x16 pooled outputs per block (conv-res tile 32x32 on the 96x192 grid)
// LDS: xd tile 40x40 (6.4KB) + conv1 tile 8x36x36 (41.5KB) + weights (7.3KB)
// conv2 inner loop amortizes the 8 weight broadcasts over 4 pool sub-pixels:
// 12 LDS reads per 32 FMAs.
// ---------------------------------------------------------------------------
__global__ __launch_bounds__(256) void stageA_kernel(
    const float* __restrict__ x,
    const float* __restrict__ c1w, const float* __restrict__ c1b,
    const float* __restrict__ c2w, const float* __restrict__ c2b,
    float* __restrict__ p1)
{
  __shared__ float s_xd[40 * 40];
  __shared__ float s_c1[8 * 36 * 36];
  __shared__ float s_w1[8 * 25];
  __shared__ float s_w2[8 * 8 * 25];
  __shared__ float s_b1v[8];
  __shared__ float s_b2v[8];

  const int tid = threadIdx.x;
  const int b    = blockIdx.y;
  const int tile = blockIdx.x;          // 0..17  (3 x 6 tiles)
  const int ty = tile / 6, tx = tile % 6;
  const int y0 = ty * 32, x0 = tx * 32; // conv-resolution origin (96x192)

  for (int i = tid; i < 200;  i += 256) s_w1[i] = c1w[i];
  for (int i = tid; i < 1600; i += 256) s_w2[i] = c2w[i];
  if (tid < 8) { s_b1v[tid] = c1b[tid]; s_b2v[tid] = c2b[tid]; }

  // ---- downsample x (2x2 mean) into s_xd, halo of 4 on the 96x192 grid ----
  const float* xb = x + (size_t)b * 192 * 384;
  for (int i = tid; i < 40 * 40; i += 256) {
    int ly = i / 40, lx = i % 40;
    int gy = y0 - 4 + ly, gx = x0 - 4 + lx;
    float v = 0.f;
    if (gy >= 0 && gy < 96 && gx >= 0 && gx < 192) {
      const float* p = xb + (size_t)(2 * gy) * 384 + 2 * gx;
      v = 0.25f * (p[0] + p[1] + p[384] + p[385]);
    }
    s_xd[i] = v;
  }
  __syncthreads();

  // ---- conv1 (1->8, 5x5 SAME), halo of 2 -> s_c1 ----
  for (int i = tid; i < 8 * 36 * 36; i += 256) {
    int ci = i / (36 * 36);
    int r  = i % (36 * 36);
    int ly = r / 36, lx = r % 36;
    int gy = y0 - 2 + ly, gx = x0 - 2 + lx;
    float v = 0.f;
    if (gy >= 0 && gy < 96 && gx >= 0 && gx < 192) {  // conv2 sees zeros outside image
      float acc = s_b1v[ci];
      const float* wp = s_w1 + ci * 25;
      #pragma unroll
      for (int ky = 0; ky < 5; ++ky)
        #pragma unroll
        for (int kx = 0; kx < 5; ++kx)
          acc += s_xd[(ly + ky) * 40 + lx + kx] * wp[ky * 5 + kx];
      v = acc;
    }
    s_c1[i] = v;
  }
  __syncthreads();

  // ---- conv2 (8->8) + relu + 2x2 maxpool : one pooled pixel per thread ----
  // acc[s][co] for the 4 pool sub-pixels kept live so each weight broadcast
  // feeds 4 FMAs.
  const int py = tid / 16, px = tid % 16;
  const int dy0 = 2 * py, dx0 = 2 * px;

  float acc[4][8];
  #pragma unroll
  for (int s = 0; s < 4; ++s)
    #pragma unroll
    for (int co = 0; co < 8; ++co) acc[s][co] = s_b2v[co];

  for (int ci = 0; ci < 8; ++ci) {
    const float* cin = s_c1 + ci * (36 * 36);
    #pragma unroll
    for (int ky = 0; ky < 5; ++ky) {
      #pragma unroll
      for (int kx = 0; kx < 5; ++kx) {
        float w[8];
        #pragma unroll
        for (int co = 0; co < 8; ++co)
          w[co] = s_w2[(co * 8 + ci) * 25 + ky * 5 + kx];
        #pragma unroll
        for (int s = 0; s < 4; ++s) {
          float v = cin[(dy0 + (s >> 1) + ky) * 36 + dx0 + (s & 1) + kx];
          #pragma unroll
          for (int co = 0; co < 8; ++co) acc[s][co] += v * w[co];
        }
      }
    }
  }

  const int gpy = ty * 16 + py, gpx = tx * 16 + px;
  #pragma unroll
  for (int co = 0; co < 8; ++co) {
    float m = fmaxf(fmaxf(acc[0][co], acc[1][co]),
                    fmaxf(acc[2][co], acc[3][co]));
    p1[(((size_t)b * 8 + co) * 48 + gpy) * 96 + gpx] = fmaxf(m, 0.f);
  }
}

// ---------------------------------------------------------------------------
// Stage B1: conv3 (8->8, 5x5 SAME, no relu) on [256,8,48,96] -> c3 buffer
// Tile: 16x32 outputs; LDS input 8x20x36 (23KB) + weights (6.4KB)
// Each thread carries 2 output pixels x 8 co: 10 LDS reads per 16 FMAs.
// ---------------------------------------------------------------------------
__global__ __launch_bounds__(256) void convB1_kernel(
    const float* __restrict__ p1,
    const float* __restrict__ c3w, const float* __restrict__ c3b,
    float* __restrict__ c3)
{
  __shared__ float s_in[8 * 20 * 36];
  __shared__ float s_w[1600];
  __shared__ float s_bv[8];

  const int tid = threadIdx.x;
  const int b    = blockIdx.y;
  const int tile = blockIdx.x;           // 0..8 (3 x 3)
  const int tyi = tile / 3, txi = tile % 3;
  const int oy0 = tyi * 16, ox0 = txi * 32;

  for (int i = tid; i < 1600; i += 256) s_w[i] = c3w[i];
  if (tid < 8) s_bv[tid] = c3b[tid];

  const float* pb = p1 + (size_t)b * 8 * 48 * 96;
  for (int i = tid; i < 8 * 20 * 36; i += 256) {
    int ci = i / (20 * 36);
    int r  = i % (20 * 36);
    int ly = r / 36, lx = r % 36;
    int gy = oy0 - 2 + ly, gx = ox0 - 2 + lx;
    float v = 0.f;
    if (gy >= 0 && gy < 48 && gx >= 0 && gx < 96)
      v = pb[(ci * 48 + gy) * 96 + gx];
    s_in[i] = v;
  }
  __syncthreads();

  // two pixels per thread: p0 = tid, p1 = tid + 256   (512 = 16x32 pixels)
  const int oyA = tid / 32,        oxA = tid % 32;
  const int oyB = (tid + 256) / 32, oxB = (tid + 256) % 32;

  float acc[2][8];
  #pragma unroll
  for (int co = 0; co < 8; ++co) { acc[0][co] = s_bv[co]; acc[1][co] = s_bv[co]; }

  for (int ci = 0; ci < 8; ++ci) {
    const float* cin = s_in + ci * 720;
    #pragma unroll
    for (int ky = 0; ky < 5; ++ky) {
      #pragma unroll
      for (int kx = 0; kx < 5; ++kx) {
        float w[8];
        #pragma unroll
        for (int co = 0; co < 8; ++co)
          w[co] = s_w[(co * 8 + ci) * 25 + ky * 5 + kx];
        float vA = cin[(oyA + ky) * 36 + oxA + kx];
        float vB = cin[(oyB + ky) * 36 + oxB + kx];
        #pragma unroll
        for (int co = 0; co < 8; ++co) {
          acc[0][co] += vA * w[co];
          acc[1][co] += vB * w[co];
        }
      }
    }
  }

  #pragma unroll
  for (int co = 0; co < 8; ++co) {
    c3[(((size_t)b * 8 + co) * 48 + oy0 + oyA) * 96 + ox0 + oxA] = acc[0][co];
    c3[(((size_t)b * 8 + co) * 48 + oy0 + oyB) * 96 + ox0 + oxB] = acc[1][co];
  }
}

// ---------------------------------------------------------------------------
// Stage B2: conv4 (8->8) + relu + 2x2 maxpool -> p2 (=hfeat) [256, 8, 24, 48]
// Tile: 16x16 pooled outputs (conv region 32x32); LDS 8x36x36 (41.5KB)
// Same 4-subpixel weight amortization as stage A.
// ---------------------------------------------------------------------------
__global__ __launch_bounds__(256) void convB2_kernel(
    const float* __restrict__ c3,
    const float* __restrict__ c4w, const float* __restrict__ c4b,
    float* __restrict__ p2)
{
  __shared__ float s_in[8 * 36 * 36];
  __shared__ float s_w[1600];
  __shared__ float s_bv[8];

  const int tid = threadIdx.x;
  const int b    = blockIdx.y;
  const int tile = blockIdx.x;           // 0..5 (2 x 3)
  const int tyi = tile / 3, txi = tile % 3;
  const int ppy0 = tyi * 16, ppx0 = txi * 16;
  const int cy0 = 2 * ppy0, cx0 = 2 * ppx0;

  for (int i = tid; i < 1600; i += 256) s_w[i] = c4w[i];
  if (tid < 8) s_bv[tid] = c4b[tid];

  const float* cb = c3 + (size_t)b * 8 * 48 * 96;
  for (int i = tid; i < 8 * 36 * 36; i += 256) {
    int ci = i / 1296;
    int r  = i % 1296;
    int ly = r / 36, lx = r % 36;
    int gy = cy0 - 2 + ly, gx = cx0 - 2 + lx;
    float v = 0.f;
    if (gy >= 0 && gy < 48 && gx >= 0 && gx < 96)
      v = cb[(ci * 48 + gy) * 96 + gx];
    s_in[i] = v;
  }
  __syncthreads();

  const int py = tid / 16, px = tid % 16;
  const int gpy = ppy0 + py, gpx = ppx0 + px;
  if (gpy < 24 && gpx < 48) {
    const int dy0 = 2 * py, dx0 = 2 * px;
    float acc[4][8];
    #pragma unroll
    for (int s = 0; s < 4; ++s)
      #pragma unroll
      for (int co = 0; co < 8; ++co) acc[s][co] = s_bv[co];

    for (int ci = 0; ci < 8; ++ci) {
      const float* cin = s_in + ci * 1296;
      #pragma unroll
      for (int ky = 0; ky < 5; ++ky) {
        #pragma unroll
        for (int kx = 0; kx < 5; ++kx) {
          float w[8];
          #pragma unroll
          for (int co = 0; co < 8; ++co)
            w[co] = s_w[(co * 8 + ci) * 25 + ky * 5 + kx];
          #pragma unroll
          for (int s = 0; s < 4; ++s) {
            float v = cin[(dy0 + (s >> 1) + ky) * 36 + dx0 + (s & 1) + kx];
            #pragma unroll
            for (int co = 0; co < 8; ++co) acc[s][co] += v * w[co];
          }
        }
      }
    }

    #pragma unroll
    for (int co = 0; co < 8; ++co) {
      float m = fmaxf(fmaxf(acc[0][co], acc[1][co]),
                      fmaxf(acc[2][co], acc[3][co]));
      p2[(((size_t)b * 8 + co) * 24 + gpy) * 48 + gpx] = fmaxf(m, 0.f);
    }
  }
}

// ---------------------------------------------------------------------------
// Stage M: per-feature MLP 8->32->32->1 over batch 256 using
//          V_WMMA_F32_16X16X4_F32 (exact fp32 accumulate).
// 1 block (8 wave32) per feature. M=256 (16 tiles) x N=32 (2 tiles)
// => 32 wave-tiles, 4 per wave. t1/t2 share one 32KB LDS buffer.
// Masked-out features skip everything.
// ---------------------------------------------------------------------------
__global__ __launch_bounds__(256) void mlp_wmma_kernel(
    const float* __restrict__ hfeat,    // [256, 8, 1152]
    const float* __restrict__ mask,     // [192, 384]
    const float* __restrict__ W1,       // [1152, 8, 32]
    const float* __restrict__ b1,       // [1152, 32]
    const float* __restrict__ W2,       // [1152, 32, 32]
    const float* __restrict__ b2,       // [1152, 32]
    const float* __restrict__ W3,       // [1152, 32]
    float* __restrict__ feat)           // [1152, 256]
{
  const int f  = blockIdx.x;
  const int fy = f / 48, fx = f % 48;
  if (mask[(fy * 8) * 384 + fx * 8] <= 0.5f) return;  // uniform block-wide exit

  __shared__ float s_t[256 * 32];
  __shared__ float s_w1[8 * 32];
  __shared__ float s_w2[32 * 32];
  __shared__ float s_b1[32];
  __shared__ float s_b2[32];
  __shared__ float s_w3[32];

  const int tid  = threadIdx.x;
  const int lane = tid & 31;
  const int wave = tid >> 5;

  for (int i = tid; i < 256;  i += 256) s_w1[i] = W1[(size_t)f * 256 + i];
  for (int i = tid; i < 1024; i += 256) s_w2[i] = W2[(size_t)f * 1024 + i];
  if (tid < 32) {
    s_b1[tid] = b1[(size_t)f * 32 + tid];
    s_b2[tid] = b2[(size_t)f * 32 + tid];
    s_w3[tid] = W3[(size_t)f * 32 + tid];
  }
  __syncthreads();

  // WMMA 16x16x4 f32 operand lane mapping (ISA 7.12.2):
  //  A 16x4 : lanes 0-15 -> K0/K1 in v[0]/v[1]; lanes 16-31 -> K2/K3
  //  B 4x16 : same K split across lane halves, N = lane&15
  //  C/D    : 8 VGPRs; lanes 0-15 -> M=r, lanes 16-31 -> M=8+r; N = lane&15
  const int lr    = lane & 15;
  const int kOff  = (lane < 16) ? 0 : 2;
  const int hiSel = (lane & 16) ? 8 : 0;

  // ---- layer 1: [256x8] x [8x32] + b1, relu -> s_t ----
  for (int t = wave; t < 32; t += 8) {
    const int mt = t >> 1, nt = t & 1;
    const int row = mt * 16 + lr;
    const int col = nt * 16 + lr;
    v8f acc = {0.f, 0.f, 0.f, 0.f, 0.f, 0.f, 0.f, 0.f};
    #pragma unroll
    for (int k = 0; k < 2; ++k) {
      const int c0 = 4 * k + kOff;
      v2f a, bm;
      a.x  = hfeat[((size_t)row * 8 + c0    ) * 1152 + f];
      a.y  = hfeat[((size_t)row * 8 + c0 + 1) * 1152 + f];
      bm.x = s_w1[(c0    ) * 32 + col];
      bm.y = s_w1[(c0 + 1) * 32 + col];
      acc = __builtin_amdgcn_wmma_f32_16x16x4_f32(false, a, false, bm,
                                                  (short)0, acc, false, false);
    }
    #pragma unroll
    for (int r = 0; r < 8; ++r) {
      const int drow = mt * 16 + hiSel + r;
      s_t[drow * 32 + col] = fmaxf(acc[r] + s_b1[col], 0.f);
    }
  }
  __syncthreads();

  // ---- layer 2: [256x32] x [32x32] (accumulate in regs; s_t reused) ----
  v8f acc2[4];
  int tc = 0;
  for (int t = wave; t < 32; t += 8, ++tc) {
    const int mt = t >> 1, nt = t & 1;
    const int row = mt * 16 + lr;
    const int col = nt * 16 + lr;
    v8f acc = {0.f, 0.f, 0.f, 0.f, 0.f, 0.f, 0.f, 0.f};
    #pragma unroll
    for (int k = 0; k < 8; ++k) {
      const int c0 = 4 * k + kOff;
      v2f a, bm;
      a.x  = s_t[row * 32 + c0];
      a.y  = s_t[row * 32 + c0 + 1];
      bm.x = s_w2[(c0    ) * 32 + col];
      bm.y = s_w2[(c0 + 1) * 32 + col];
      acc = __builtin_amdgcn_wmma_f32_16x16x4_f32(false, a, false, bm,
                                                  (short)0, acc, false, false);
    }
    acc2[tc] = acc;
  }
  __syncthreads();   // all reads of t1 complete before overwrite

  tc = 0;
  for (int t = wave; t < 32; t += 8, ++tc) {
    const int mt = t >> 1, nt = t & 1;
    const int col = nt * 16 + lr;
    #pragma unroll
    for (int r = 0; r < 8; ++r) {
      const int drow = mt * 16 + hiSel + r;
      s_t[drow * 32 + col] = fmaxf(acc2[tc][r] + s_b2[col], 0.f);
    }
  }
  __syncthreads();

  // ---- layer 3: dot with W3 -> feat[f][b] ----
  float sum = 0.f;
  #pragma unroll
  for (int g = 0; g < 32; ++g) sum += s_t[tid * 32 + g] * s_w3[g];
  feat[(size_t)f * 256 + tid] = sum;
}

// ---------------------------------------------------------------------------
// Final reduction: out[b] = sigmoid( sum_f mask(f)*feat[f][b] + add_bias )
// Fixed sequential feature order => bitwise deterministic.
// ---------------------------------------------------------------------------
__global__ __launch_bounds__(256) void reduce_kernel(
    const float* __restrict__ feat,
    const float* __restrict__ mask,
    const float* __restrict__ add_bias,
    float* __restrict__ out)
{
  const int b = threadIdx.x;
  float acc = add_bias[0];
  for (int f = 0; f < 1152; ++f) {
    const int fy = f / 48, fx = f % 48;
    if (mask[(fy * 8) * 384 + fx * 8] > 0.5f)
      acc += feat[(size_t)f * 256 + b];
  }
  out[b] = 1.f / (1.f + expf(-acc));
}

// ---------------------------------------------------------------------------
extern "C" void kernel_launch(void* const* d_in, const int* in_sizes, int n_in,
                              void* d_out, int out_size, void* d_ws, size_t ws_size,
                              hipStream_t stream) {
  const float* x    = (const float*)d_in[0];
  const float* mask = (const float*)d_in[1];
  const float* c1w  = (const float*)d_in[2];
  const float* c1b  = (const float*)d_in[3];
  const float* c2w  = (const float*)d_in[4];
  const float* c2b  = (const float*)d_in[5];
  const float* c3w  = (const float*)d_in[6];
  const float* c3b  = (const float*)d_in[7];
  const float* c4w  = (const float*)d_in[8];
  const float* c4b  = (const float*)d_in[9];
  const float* W1   = (const float*)d_in[10];
  const float* b1   = (const float*)d_in[11];
  const float* W2   = (const float*)d_in[12];
  const float* b2   = (const float*)d_in[13];
  const float* W3   = (const float*)d_in[14];
  const float* add_bias = (const float*)d_in[15];
  float* out = (float*)d_out;

  // workspace layout (floats)
  float* ws   = (float*)d_ws;
  float* p1   = ws;                       // 256*8*48*96 = 9437184
  float* c3   = p1 + 9437184;             // 9437184
  float* p2   = c3 + 9437184;             // 256*8*24*48 = 2359296
  float* feat = p2 + 2359296;             // 1152*256    = 294912

  dim3 blk(256);
  stageA_kernel <<<dim3(18, 256), blk, 0, stream>>>(x, c1w, c1b, c2w, c2b, p1);
  convB1_kernel <<<dim3(9, 256),  blk, 0, stream>>>(p1, c3w, c3b, c3);
  convB2_kernel <<<dim3(6, 256),  blk, 0, stream>>>(c3, c4w, c4b, p2);
  mlp_wmma_kernel<<<dim3(1152),   blk, 0, stream>>>(p2, mask, W1, b1, W2, b2, W3, feat);
  reduce_kernel  <<<dim3(1),      blk, 0, stream>>>(feat, mask, add_bias, out);
}